// SAGEConv_6201932775989
// MI455X (gfx1250) — compile-verified
//
#include <hip/hip_runtime.h>
#include <stdint.h>

#define N_NODES 100000
#define N_EDGES 1600000
#define D_FEAT  32

typedef __attribute__((ext_vector_type(4))) float v4f;   // native vector: OK for NT builtins

// ---------------------------------------------------------------------------
// Kernel 1: zero the sums accumulator (d_out) and degree buffer (d_ws).
// ---------------------------------------------------------------------------
__global__ void sage_zero_kernel(float4* __restrict__ out4, float4* __restrict__ deg4) {
    const int i = blockIdx.x * blockDim.x + threadIdx.x;
    const float4 z = make_float4(0.f, 0.f, 0.f, 0.f);
    if (i < (N_NODES * D_FEAT) / 4) out4[i] = z;           // 800,000 float4
    if (i < N_NODES / 4)            deg4[i] = z;           // 25,000 float4
}

// ---------------------------------------------------------------------------
// Kernel 2: edge scatter. 8 lanes per edge, 32 edges per 256-thread block
// iteration. The sequential `pattern` stream (204.8 MB, dominant HBM traffic)
// is double-buffered through LDS with GLOBAL_LOAD_ASYNC_TO_LDS_B128
// (non-temporal, so it doesn't evict the L2-resident rel/sums working set),
// overlapped via ASYNCcnt: issue tile t+1, wait asynccnt<=1, consume tile t.
// rel[src] gathers hit L2 (12.8 MB table << 192 MB). Accumulation is
// global_atomic_add_f32 resolved in L2 (sums also L2-resident).
// ---------------------------------------------------------------------------
__device__ __forceinline__ void async_pat_load(unsigned ldsAddr, const float* gsrc) {
    asm volatile("global_load_async_to_lds_b128 %0, %1, off th:TH_LOAD_NT"
                 :: "v"(ldsAddr), "v"(gsrc)
                 : "memory");
}

__global__ void __launch_bounds__(256)
sage_scatter_kernel(const float* __restrict__ rel,
                    const float* __restrict__ pattern,
                    const int*   __restrict__ src,
                    const int*   __restrict__ dst,
                    float*       __restrict__ sums,
                    float*       __restrict__ deg) {
    __shared__ float4 patTile[2][256];                     // 2 x 4KB staging buffers

    const int tid    = threadIdx.x;
    const int eLocal = tid >> 3;                           // edge within tile [0,32)
    const int sub    = tid & 7;                            // float4 chunk within row [0,8)

    // LDS byte addresses of this thread's private staging slots. Generic->LDS
    // addresses carry the LDS offset in addr[31:0] (aperture in high bits).
    const unsigned ldsSlot[2] = {
        (unsigned)(uintptr_t)&patTile[0][tid],
        (unsigned)(uintptr_t)&patTile[1][tid]
    };

    const int tileCount = N_EDGES / 32;                    // 50,000 (exact)
    int tile = blockIdx.x;

    // Prime the pipeline: issue the first tile's stream load.
    if (tile < tileCount)
        async_pat_load(ldsSlot[0], pattern + (size_t)tile * (32 * D_FEAT) + tid * 4);

    int buf = 0;
    for (; tile < tileCount; tile += gridDim.x) {
        const int next = tile + gridDim.x;
        if (next < tileCount) {
            // Issue next tile into the other buffer, then wait for current
            // (async loads complete in order: <=1 outstanding => current done).
            async_pat_load(ldsSlot[buf ^ 1],
                           pattern + (size_t)next * (32 * D_FEAT) + tid * 4);
            asm volatile("s_wait_asynccnt 0x1" ::: "memory");
        } else {
            asm volatile("s_wait_asynccnt 0x0" ::: "memory");
        }

        const int e = tile * 32 + eLocal;
        const int s = __builtin_nontemporal_load(src + e); // index stream: NT
        const int d = __builtin_nontemporal_load(dst + e);

        const float4 pw = patTile[buf][tid];               // ds_load_b128
        const float4 rv = *(reinterpret_cast<const float4*>(rel) + ((size_t)s << 3) + sub);

        float* sb = sums + ((size_t)d << 5) + (sub << 2);
        atomicAdd(sb + 0, rv.x * pw.x);
        atomicAdd(sb + 1, rv.y * pw.y);
        atomicAdd(sb + 2, rv.z * pw.z);
        atomicAdd(sb + 3, rv.w * pw.w);
        if (sub == 0) atomicAdd(deg + d, 1.0f);

        buf ^= 1;
    }
}

// ---------------------------------------------------------------------------
// Kernel 3: out = sums / max(deg, 1) + rel   (vectorized, memory bound)
// ---------------------------------------------------------------------------
__global__ void sage_finalize_kernel(const v4f* __restrict__ rel4,
                                     const float* __restrict__ deg,
                                     v4f* __restrict__ out4) {
    const int i = blockIdx.x * blockDim.x + threadIdx.x;
    if (i < (N_NODES * D_FEAT) / 4) {
        const float dg = fmaxf(deg[i >> 3], 1.0f);         // 8 float4 per node row
        v4f s = out4[i];
        const v4f r = __builtin_nontemporal_load(rel4 + i); // last use of rel
        s.x = s.x / dg + r.x;
        s.y = s.y / dg + r.y;
        s.z = s.z / dg + r.z;
        s.w = s.w / dg + r.w;
        out4[i] = s;
    }
}

extern "C" void kernel_launch(void* const* d_in, const int* in_sizes, int n_in,
                              void* d_out, int out_size, void* d_ws, size_t ws_size,
                              hipStream_t stream) {
    const float* rel     = (const float*)d_in[0];          // [N_NODES, 32] f32
    const float* pattern = (const float*)d_in[1];          // [N_EDGES, 32] f32
    const int*   src     = (const int*)d_in[2];            // [N_EDGES] int
    const int*   dst     = (const int*)d_in[3];            // [N_EDGES] int

    float* sums = (float*)d_out;                           // N_NODES*32 f32 accumulator -> final out
    float* deg  = (float*)d_ws;                            // N_NODES f32 degree counts

    (void)in_sizes; (void)n_in; (void)out_size; (void)ws_size;

    // 1) zero accumulators (d_out/d_ws are poisoned by the harness)
    const int zeroElems = (N_NODES * D_FEAT) / 4;          // 800,000 (covers deg too)
    sage_zero_kernel<<<(zeroElems + 255) / 256, 256, 0, stream>>>(
        (float4*)sums, (float4*)deg);

    // 2) edge scatter: 50,000 tiles of 32 edges, grid-stride over 8192 blocks
    sage_scatter_kernel<<<8192, 256, 0, stream>>>(rel, pattern, src, dst, sums, deg);

    // 3) finalize mean + residual
    sage_finalize_kernel<<<(zeroElems + 255) / 256, 256, 0, stream>>>(
        (const v4f*)rel, deg, (v4f*)sums);
}